// WeightedMultiHeadAttentionLayer_59596966199896
// MI455X (gfx1250) — compile-verified
//
#include <hip/hip_runtime.h>
#include <hip/hip_bf16.h>

typedef __attribute__((ext_vector_type(2))) float v2f;
typedef __attribute__((ext_vector_type(8))) float v8f;

#define HD 64          // H*D
#define HEADS 8
#define DDIM 8
#define TILE 16        // rows per wave tile
#define WAVES 8        // waves per block in GEMM kernels
#define WPAD 66        // padded LDS stride for transposed W (2-way conflict max)
#define INV_SQRT_D 0.35355339059327373f   // 1/sqrt(8)
#define CLAMP_V 5.0f

// Stage W[64][64] into LDS transposed: Wt[col][k], stride WPAD.
__device__ __forceinline__ void stage_w_transposed(const float* __restrict__ W,
                                                   float* __restrict__ Wt) {
    for (int i = threadIdx.x; i < HD * HD; i += blockDim.x) {
        const int r = i >> 6;      // k
        const int c = i & 63;      // col
        Wt[c * WPAD + r] = W[i];
    }
}

// One wave: 16-row x 64-col f32 GEMM tile via V_WMMA_F32_16X16X4_F32.
// A from global (b64/lane/K-step), B from LDS-transposed W (ds_load_b64),
// bias pre-loaded into the accumulators (C-layout: lane's col is constant).
__device__ __forceinline__ void wmma_tile_16x64(
    const float* __restrict__ ap,    // row pointer for this lane's A row
    const float* __restrict__ Wt,    // LDS transposed weights
    const float* __restrict__ b,     // bias
    int n, int hi,
    v8f& acc0, v8f& acc1, v8f& acc2, v8f& acc3) {
    const float b0 = b[n], b1 = b[16 + n], b2 = b[32 + n], b3 = b[48 + n];
    acc0 = (v8f){b0, b0, b0, b0, b0, b0, b0, b0};
    acc1 = (v8f){b1, b1, b1, b1, b1, b1, b1, b1};
    acc2 = (v8f){b2, b2, b2, b2, b2, b2, b2, b2};
    acc3 = (v8f){b3, b3, b3, b3, b3, b3, b3, b3};
    #pragma unroll
    for (int k = 0; k < HD; k += 4) {
        const int kk = k + 2 * hi;
        v2f A;
        A.x = ap[kk + 0];
        A.y = ap[kk + 1];
        const v2f B0 = *(const v2f*)&Wt[(0 * 16 + n) * WPAD + kk];
        const v2f B1 = *(const v2f*)&Wt[(1 * 16 + n) * WPAD + kk];
        const v2f B2 = *(const v2f*)&Wt[(2 * 16 + n) * WPAD + kk];
        const v2f B3 = *(const v2f*)&Wt[(3 * 16 + n) * WPAD + kk];
        acc0 = __builtin_amdgcn_wmma_f32_16x16x4_f32(false, A, false, B0, (short)0, acc0, false, false);
        acc1 = __builtin_amdgcn_wmma_f32_16x16x4_f32(false, A, false, B1, (short)0, acc1, false, false);
        acc2 = __builtin_amdgcn_wmma_f32_16x16x4_f32(false, A, false, B2, (short)0, acc2, false, false);
        acc3 = __builtin_amdgcn_wmma_f32_16x16x4_f32(false, A, false, B3, (short)0, acc3, false, false);
    }
}

// ---------------------------------------------------------------------------
// Node projection: out[N,64] = X[N,64] @ W[64,64] + b.
// 256-thread block = 8 waves, each owning one 16-row tile; W staged once.
// ---------------------------------------------------------------------------
__global__ __launch_bounds__(256)
void proj_kernel(const float* __restrict__ X,
                 const float* __restrict__ W,
                 const float* __restrict__ b,
                 float* __restrict__ out, int nrows) {
    __shared__ float Wt[HD * WPAD];
    stage_w_transposed(W, Wt);
    __syncthreads();

    const int wave = threadIdx.x >> 5;
    const int lane = threadIdx.x & 31;
    const int n    = lane & 15;
    const int hi   = lane >> 4;
    const int row0 = (blockIdx.x * WAVES + wave) * TILE;
    if (row0 >= nrows) return;                 // wave-uniform exit

    const int arow = min(row0 + n, nrows - 1);
    v8f acc0, acc1, acc2, acc3;
    wmma_tile_16x64(X + (size_t)arow * HD, Wt, b, n, hi, acc0, acc1, acc2, acc3);

    #pragma unroll
    for (int i = 0; i < 8; ++i) {
        const int row = row0 + i + 8 * hi;
        if (row < nrows) {
            float* __restrict__ op = out + (size_t)row * HD;
            op[n]      = acc0[i];
            op[16 + n] = acc1[i];
            op[32 + n] = acc2[i];
            op[48 + n] = acc3[i];
        }
    }
}

// ---------------------------------------------------------------------------
// Fused edge kernel: per-wave Eh tile = edge_attr@WE+bE kept in LDS (never
// hits HBM), then per (edge,head): s = clamp(sum_d K*Q*E / sqrt(D)), ex=exp(s)
// stored to ws, atomic-add into den[dst,h]. Clamp to +-5 makes the segment-max
// shift cancel exactly, so no max pass is needed.
// Same-wave LDS is in-order (DScnt) => no barrier between E write and read.
// ---------------------------------------------------------------------------
__global__ __launch_bounds__(256)
void edge_score_kernel(const float* __restrict__ eattr,
                       const float* __restrict__ W,
                       const float* __restrict__ b,
                       const float* __restrict__ Kh,
                       const float* __restrict__ Qh,
                       const int*   __restrict__ src,
                       const int*   __restrict__ dst,
                       float* __restrict__ exs,
                       float* __restrict__ den, int nedges) {
    __shared__ float Wt[HD * WPAD];
    __shared__ float Etile[WAVES][TILE * HD];
    stage_w_transposed(W, Wt);
    __syncthreads();

    const int wave = threadIdx.x >> 5;
    const int lane = threadIdx.x & 31;
    const int n    = lane & 15;
    const int hi   = lane >> 4;
    const int e0   = (blockIdx.x * WAVES + wave) * TILE;
    if (e0 >= nedges) return;                  // wave-uniform exit

    const int arow = min(e0 + n, nedges - 1);
    v8f acc0, acc1, acc2, acc3;
    wmma_tile_16x64(eattr + (size_t)arow * HD, Wt, b, n, hi, acc0, acc1, acc2, acc3);

    float* __restrict__ Et = Etile[wave];
    #pragma unroll
    for (int i = 0; i < 8; ++i) {
        const int r = i + 8 * hi;              // local edge row within tile
        Et[r * HD + n]      = acc0[i];
        Et[r * HD + 16 + n] = acc1[i];
        Et[r * HD + 32 + n] = acc2[i];
        Et[r * HD + 48 + n] = acc3[i];
    }

    // 16 edges x 8 heads = 128 tasks, 4 per lane (same-wave LDS -> in order)
    #pragma unroll
    for (int t = lane; t < TILE * HEADS; t += 32) {
        const int e  = t >> 3;
        const int h  = t & 7;
        const int ge = e0 + e;
        if (ge < nedges) {
            const int s = src[ge];
            const int d = dst[ge];
            const float* __restrict__ kp = Kh + (size_t)s * HD + h * DDIM;
            const float* __restrict__ qp = Qh + (size_t)d * HD + h * DDIM;
            const float* __restrict__ ep = Et + e * HD + h * DDIM;
            float sum = 0.f;
            #pragma unroll
            for (int dd = 0; dd < DDIM; ++dd) sum += kp[dd] * qp[dd] * ep[dd];
            sum *= INV_SQRT_D;
            sum = fminf(CLAMP_V, fmaxf(-CLAMP_V, sum));
            const float ex = __expf(sum);
            exs[(size_t)ge * HEADS + h] = ex;
            unsafeAtomicAdd(&den[(size_t)d * HEADS + h], ex);
        }
    }
}

// ---------------------------------------------------------------------------
// Scatter: out[dst, c] += Vh[src, c] * coeff * ex[e,h]/(den[dst,h]+1e-16)
// one thread per (edge, channel); native f32 atomics land in the 192MB L2.
// ---------------------------------------------------------------------------
__global__ __launch_bounds__(256)
void aggregate_kernel(const float* __restrict__ Vh,
                      const int*   __restrict__ src,
                      const int*   __restrict__ dst,
                      const float* __restrict__ exs,
                      const float* __restrict__ den,
                      float coeff, float* __restrict__ out, long total) {
    const long idx = (long)blockIdx.x * blockDim.x + threadIdx.x;
    if (idx >= total) return;
    const int e = (int)(idx >> 6);
    const int c = (int)(idx & 63);
    const int h = c >> 3;
    const int s = src[e];
    const int d = dst[e];
    const float score = exs[(size_t)e * HEADS + h] /
                        (den[(size_t)d * HEADS + h] + 1e-16f) * coeff;
    unsafeAtomicAdd(&out[(size_t)d * HD + c], Vh[(size_t)s * HD + c] * score);
}

// ---------------------------------------------------------------------------
extern "C" void kernel_launch(void* const* d_in, const int* in_sizes, int n_in,
                              void* d_out, int out_size, void* d_ws, size_t ws_size,
                              hipStream_t stream) {
    const float* x    = (const float*)d_in[0];
    const float* ea   = (const float*)d_in[1];
    const float* eaf  = (const float*)d_in[2];
    const int*   ei   = (const int*)d_in[3];   // [2,E]
    const int*   eif  = (const int*)d_in[4];
    const float* WQ  = (const float*)d_in[5];  const float* bQ  = (const float*)d_in[6];
    const float* WK  = (const float*)d_in[7];  const float* bK  = (const float*)d_in[8];
    const float* WE  = (const float*)d_in[9];  const float* bE  = (const float*)d_in[10];
    const float* WQf = (const float*)d_in[11]; const float* bQf = (const float*)d_in[12];
    const float* WKf = (const float*)d_in[13]; const float* bKf = (const float*)d_in[14];
    const float* WEf = (const float*)d_in[15]; const float* bEf = (const float*)d_in[16];
    const float* WV  = (const float*)d_in[17]; const float* bV  = (const float*)d_in[18];
    float* out = (float*)d_out;

    const int N = in_sizes[0] / HD;
    const int E = in_sizes[1] / HD;

    // workspace carve-up (floats)
    float* ws   = (float*)d_ws;
    float* Qh   = ws;
    float* Kh   = Qh  + (size_t)N * HD;
    float* Vh   = Kh  + (size_t)N * HD;
    float* Qhf  = Vh  + (size_t)N * HD;
    float* Khf  = Qhf + (size_t)N * HD;
    float* exr  = Khf + (size_t)N * HD;       // [E,H]
    float* exf  = exr + (size_t)E * HEADS;    // [E,H]
    float* denr = exf + (size_t)E * HEADS;    // [N,H]
    float* denf = denr + (size_t)N * HEADS;   // [N,H]

    hipMemsetAsync(out,  0, (size_t)N * HD * sizeof(float), stream);
    hipMemsetAsync(denr, 0, 2 * (size_t)N * HEADS * sizeof(float), stream);

    const int ntiles  = (N + TILE - 1) / TILE;
    const int nblocks = (ntiles + WAVES - 1) / WAVES;
    proj_kernel<<<nblocks, 256, 0, stream>>>(x, WQ,  bQ,  Qh,  N);
    proj_kernel<<<nblocks, 256, 0, stream>>>(x, WK,  bK,  Kh,  N);
    proj_kernel<<<nblocks, 256, 0, stream>>>(x, WV,  bV,  Vh,  N);
    proj_kernel<<<nblocks, 256, 0, stream>>>(x, WQf, bQf, Qhf, N);
    proj_kernel<<<nblocks, 256, 0, stream>>>(x, WKf, bKf, Khf, N);

    const int etiles  = (E + TILE - 1) / TILE;
    const int eblocks = (etiles + WAVES - 1) / WAVES;
    edge_score_kernel<<<eblocks, 256, 0, stream>>>(ea,  WE,  bE,  Kh,  Qh,  ei,  ei  + E, exr, denr, E);
    edge_score_kernel<<<eblocks, 256, 0, stream>>>(eaf, WEf, bEf, Khf, Qhf, eif, eif + E, exf, denf, E);

    const long total = (long)E * HD;
    const int ablocks = (int)((total + 255) / 256);
    const float g = 0.5f;
    aggregate_kernel<<<ablocks, 256, 0, stream>>>(Vh, ei,  ei  + E, exr, denr, 1.0f / (g + 1.0f), out, total);
    aggregate_kernel<<<ablocks, 256, 0, stream>>>(Vh, eif, eif + E, exf, denf, g / (g + 1.0f),    out, total);
}